// GAT_10307921511004
// MI455X (gfx1250) — compile-verified
//
#include <hip/hip_runtime.h>
#include <math.h>
#include <stdint.h>

typedef __attribute__((ext_vector_type(2))) float v2f;
typedef __attribute__((ext_vector_type(8))) float v8f;

#define NNODES 10000
#define NH 8
#define DD 128
#define KK 16
#define LN_EPS 1e-5f
#define NTILES 625              // 10000 / 16
#define TILES_PER_BLOCK 8       // 8 waves of 32 per block

// XOR swizzle: spreads lanes (which differ in d) across LDS bank pairs while
// keeping each lane's {f, f+1} fragment contiguous (only bits >=2 are XORed,
// f is always even at fragment granularity; 4-float chunks also stay intact).
#define SWZ(d, f) ((d) * DD + ((f) ^ (((d) & 15) << 2)))

// ---------------------------------------------------------------------------
// h[head, tile*16 : +16, 0:128] = X_tile(16x128) @ W[head]^T via
// V_WMMA_F32_16X16X4_F32.  W[head] (64 KB) staged in LDS with async
// global->LDS b128 copies (ASYNCcnt), shared by 8 waves.
// A (16x4, f32): lane<16 -> M=lane, K={k,k+1}; lane>=16 -> M=lane-16, K={k+2,k+3}
// B (4x16, f32): lanes0-15 hold K-rows {k,k+1}, lanes16-31 hold {k+2,k+3}; col=lane&15
// C (16x16, f32, 8 VGPRs): elem j at lane L -> row = j + 8*(L>>4), col = L&15
// ---------------------------------------------------------------------------
__global__ void __launch_bounds__(256)
gat_transform(const float* __restrict__ x,   // N x 128
              const float* __restrict__ W,   // H x 128 x 128 (d,f)
              float* __restrict__ h_out)     // H x N x 128
{
    __shared__ float Wlds[DD * DD];          // exactly 64 KB

    const int head = blockIdx.y;
    const int t    = threadIdx.x;            // 0..255

    // ---- async swizzled copy of W[head] into LDS (b128 granularity) ----
    {
        const float* __restrict__ Wh = W + (size_t)head * DD * DD;
#pragma unroll
        for (int i = t; i < DD * (DD / 4); i += 256) {   // 4096 float4 chunks
            const int d = i >> 5;                        // row (output dim)
            const int c = i & 31;                        // float4 index in row
            const float* gptr = Wh + (size_t)d * DD + 4 * c;
            uint32_t lds_off = (uint32_t)(uintptr_t)&Wlds[SWZ(d, 4 * c)];
            asm volatile("global_load_async_to_lds_b128 %0, %1, off"
                         :: "v"(lds_off), "v"(gptr) : "memory");
        }
        // wait for this wave's async copies, then publish to all waves
        asm volatile("s_wait_asynccnt 0x0" ::: "memory");
    }
    __syncthreads();

    const int wave = t >> 5;
    const int lane = t & 31;
    const int tile = blockIdx.x * TILES_PER_BLOCK + wave;
    if (tile >= NTILES) return;

    const int half = lane >> 4;              // 0 or 1
    const int l    = lane & 15;
    const int n0   = tile * 16;

    const float* __restrict__ xrow = x + (size_t)(n0 + l) * DD + 2 * half;

    v8f acc[8];
#pragma unroll
    for (int dt = 0; dt < 8; ++dt) acc[dt] = (v8f){0,0,0,0,0,0,0,0};

    for (int k = 0; k < DD; k += 4) {
        // A fragment (global, one b64 per lane per k-step; tile read once total)
        v2f A = *(const v2f*)(xrow + k);

        // Batch all 8 B fragments from LDS into distinct registers first,
        // so the compiler issues grouped ds-waits instead of per-WMMA stalls.
        v2f Bf[8];
#pragma unroll
        for (int dt = 0; dt < 8; ++dt) {
            const int d = dt * 16 + l;
            Bf[dt] = *(const v2f*)&Wlds[SWZ(d, k + 2 * half)];
        }
#pragma unroll
        for (int dt = 0; dt < 8; ++dt) {
            acc[dt] = __builtin_amdgcn_wmma_f32_16x16x4_f32(
                false, A, false, Bf[dt], (short)0, acc[dt], false, false);
        }
    }

    // ---- store: 16 consecutive d per lane group -> coalesced rows ----
    float* __restrict__ out = h_out + ((size_t)head * NNODES + n0) * DD;
#pragma unroll
    for (int dt = 0; dt < 8; ++dt) {
#pragma unroll
        for (int j = 0; j < 8; ++j) {
            const int row = j + 8 * half;
            out[(size_t)row * DD + dt * 16 + l] = acc[dt][j];
        }
    }
}

// ---------------------------------------------------------------------------
// s_src[h,n] = h[h,n,:] . a[h, 0:128];  s_dst[h,n] = h[h,n,:] . a[h, 128:256]
// ---------------------------------------------------------------------------
__global__ void score_kernel(const float* __restrict__ h,
                             const float* __restrict__ a,   // H x 256
                             float* __restrict__ s_src,
                             float* __restrict__ s_dst)
{
    const int idx = blockIdx.x * blockDim.x + threadIdx.x;
    if (idx >= NH * NNODES) return;
    const int head = idx / NNODES;
    const int n    = idx % NNODES;
    const float* __restrict__ hv = h + ((size_t)head * NNODES + n) * DD;
    const float* __restrict__ av = a + (size_t)head * 2 * DD;
    float ss = 0.f, sd = 0.f;
#pragma unroll 8
    for (int d = 0; d < DD; ++d) {
        const float v = hv[d];
        ss += v * av[d];
        sd += v * av[DD + d];
    }
    s_src[idx] = ss;
    s_dst[idx] = sd;
}

// ---------------------------------------------------------------------------
// Per node: e[h,k] = lrelu(s_src[h,n] + s_dst[h,nbr[k]]), softmax over k,
// out[h,n,:] = sum_k alpha * h[h,nbr[k],:]; emb = elu(mean over heads)
// Block = 128 threads (threads == (head,k) for scores, == d for aggregation)
// ---------------------------------------------------------------------------
__global__ void aggregate_kernel(const float* __restrict__ h,
                                 const float* __restrict__ s_src,
                                 const float* __restrict__ s_dst,
                                 const int*   __restrict__ nbr,   // N x 16
                                 float* __restrict__ emb)         // N x 128
{
    __shared__ float alpha_s[NH][KK];
    __shared__ int   nbr_s[KK];

    const int n = blockIdx.x;
    const int t = threadIdx.x;

    if (t < KK) nbr_s[t] = nbr[(size_t)n * KK + t];
    __syncthreads();

    // ---- scores + softmax: thread t -> head = t/16, k = t%16 ----
    {
        const int hd = t >> 4;
        const int k  = t & 15;
        float e = s_src[(size_t)hd * NNODES + n] +
                  s_dst[(size_t)hd * NNODES + nbr_s[k]];
        e = (e > 0.f) ? e : 0.01f * e;                 // leaky_relu(0.01)
        float m = e;
#pragma unroll
        for (int off = 8; off >= 1; off >>= 1)
            m = fmaxf(m, __shfl_xor(m, off, 16));
        float ex = expf(e - m);
        float sum = ex;
#pragma unroll
        for (int off = 8; off >= 1; off >>= 1)
            sum += __shfl_xor(sum, off, 16);
        alpha_s[hd][k] = ex / sum;
    }
    __syncthreads();

    // ---- aggregation: thread t owns dim d = t ----
    float accd = 0.f;
#pragma unroll
    for (int hd = 0; hd < NH; ++hd) {
        const float* __restrict__ hb = h + (size_t)hd * NNODES * DD;
        float ah = 0.f;
#pragma unroll
        for (int k = 0; k < KK; ++k)
            ah += alpha_s[hd][k] * hb[(size_t)nbr_s[k] * DD + t];
        accd += ah;
    }
    accd *= (1.0f / (float)NH);                        // mean over heads
    emb[(size_t)n * DD + t] = (accd > 0.f) ? accd : expm1f(accd);  // elu
}

// ---------------------------------------------------------------------------
// g[d] = mean over nodes of emb[n,d]
// ---------------------------------------------------------------------------
__global__ void mean_nodes(const float* __restrict__ emb, float* __restrict__ g)
{
    const int d = threadIdx.x;  // 128 threads
    float s = 0.f;
    for (int n = 0; n < NNODES; ++n) s += emb[(size_t)n * DD + d];
    g[d] = s * (1.0f / (float)NNODES);
}

// ---------------------------------------------------------------------------
// MLP on the pooled 128-vector. Single block of 128 threads.
// ---------------------------------------------------------------------------
__device__ __forceinline__ float gelu_exact(float x) {
    return 0.5f * x * (1.0f + erff(x * 0.70710678118654752f));
}

__global__ void mlp_kernel(const float* __restrict__ g,
                           const float* __restrict__ ln1_g, const float* __restrict__ ln1_b,
                           const float* __restrict__ lin1_W,                      // 64x128
                           const float* __restrict__ ln2_g, const float* __restrict__ ln2_b,
                           const float* __restrict__ lin2_W, const float* __restrict__ lin2_b, // 32x64
                           const float* __restrict__ ln3_g, const float* __restrict__ ln3_b,
                           const float* __restrict__ lin3_W, const float* __restrict__ lin3_b, // 8x32
                           const float* __restrict__ lin4_W, const float* __restrict__ lin4_b, // 1x8
                           float* __restrict__ out)
{
    __shared__ float b128[128];
    __shared__ float b64[64];
    __shared__ float stat[2];
    const int t = threadIdx.x;

    // ---- LN1 over 128 ----
    b128[t] = g[t];
    __syncthreads();
    if (t == 0) {
        float m = 0.f;
        for (int i = 0; i < 128; ++i) m += b128[i];
        m *= (1.0f / 128.0f);
        float v = 0.f;
        for (int i = 0; i < 128; ++i) { float d = b128[i] - m; v += d * d; }
        v *= (1.0f / 128.0f);
        stat[0] = m; stat[1] = 1.0f / sqrtf(v + LN_EPS);
    }
    __syncthreads();
    b128[t] = (b128[t] - stat[0]) * stat[1] * ln1_g[t] + ln1_b[t];
    __syncthreads();

    // ---- lin1 (no bias) + gelu -> 64 ----
    if (t < 64) {
        float z = 0.f;
        for (int d = 0; d < 128; ++d) z += b128[d] * lin1_W[t * 128 + d];
        b64[t] = gelu_exact(z);
    }
    __syncthreads();

    // ---- LN2 over 64 ----
    if (t == 0) {
        float m = 0.f;
        for (int i = 0; i < 64; ++i) m += b64[i];
        m *= (1.0f / 64.0f);
        float v = 0.f;
        for (int i = 0; i < 64; ++i) { float d = b64[i] - m; v += d * d; }
        v *= (1.0f / 64.0f);
        stat[0] = m; stat[1] = 1.0f / sqrtf(v + LN_EPS);
    }
    __syncthreads();
    if (t < 64) b64[t] = (b64[t] - stat[0]) * stat[1] * ln2_g[t] + ln2_b[t];
    __syncthreads();

    // ---- lin2 + bias + gelu -> 32 (store in b128[0:32]) ----
    if (t < 32) {
        float z = lin2_b[t];
        for (int d = 0; d < 64; ++d) z += b64[d] * lin2_W[t * 64 + d];
        b128[t] = gelu_exact(z);
    }
    __syncthreads();

    // ---- LN3 over 32 ----
    if (t == 0) {
        float m = 0.f;
        for (int i = 0; i < 32; ++i) m += b128[i];
        m *= (1.0f / 32.0f);
        float v = 0.f;
        for (int i = 0; i < 32; ++i) { float d = b128[i] - m; v += d * d; }
        v *= (1.0f / 32.0f);
        stat[0] = m; stat[1] = 1.0f / sqrtf(v + LN_EPS);
    }
    __syncthreads();
    if (t < 32) b128[t] = (b128[t] - stat[0]) * stat[1] * ln3_g[t] + ln3_b[t];
    __syncthreads();

    // ---- lin3 + bias + gelu -> 8 (store in b64[0:8]) ----
    if (t < 8) {
        float z = lin3_b[t];
        for (int d = 0; d < 32; ++d) z += b128[d] * lin3_W[t * 32 + d];
        b64[t] = gelu_exact(z);
    }
    __syncthreads();

    // ---- lin4 + bias + relu -> scalar ----
    if (t == 0) {
        float z = lin4_b[0];
        for (int o = 0; o < 8; ++o) z += b64[o] * lin4_W[o];
        out[0] = fmaxf(z, 0.f);
    }
}

// ---------------------------------------------------------------------------
extern "C" void kernel_launch(void* const* d_in, const int* in_sizes, int n_in,
                              void* d_out, int out_size, void* d_ws, size_t ws_size,
                              hipStream_t stream)
{
    const float* node_features = (const float*)d_in[0];
    const int*   neighbours    = (const int*)  d_in[1];
    const float* Ws[3] = { (const float*)d_in[2], (const float*)d_in[4], (const float*)d_in[6] };
    const float* As[3] = { (const float*)d_in[3], (const float*)d_in[5], (const float*)d_in[7] };
    const float* ln1_g  = (const float*)d_in[8];
    const float* ln1_b  = (const float*)d_in[9];
    const float* lin1_W = (const float*)d_in[10];
    const float* ln2_g  = (const float*)d_in[11];
    const float* ln2_b  = (const float*)d_in[12];
    const float* lin2_W = (const float*)d_in[13];
    const float* lin2_b = (const float*)d_in[14];
    const float* ln3_g  = (const float*)d_in[15];
    const float* ln3_b  = (const float*)d_in[16];
    const float* lin3_W = (const float*)d_in[17];
    const float* lin3_b = (const float*)d_in[18];
    const float* lin4_W = (const float*)d_in[19];
    const float* lin4_b = (const float*)d_in[20];

    // workspace layout (floats)
    float* ws    = (float*)d_ws;
    float* h_buf = ws;                                  // H*N*D = 10,240,000
    float* s_src = h_buf + (size_t)NH * NNODES * DD;    // 80,000
    float* s_dst = s_src + (size_t)NH * NNODES;         // 80,000
    float* emb_a = s_dst + (size_t)NH * NNODES;         // N*D
    float* emb_b = emb_a + (size_t)NNODES * DD;         // N*D
    float* gvec  = emb_b + (size_t)NNODES * DD;         // 128

    const float* cur = node_features;
    float* nxt = emb_a;

    const int gx = (NTILES + TILES_PER_BLOCK - 1) / TILES_PER_BLOCK;  // 79

    for (int L = 0; L < 3; ++L) {
        gat_transform<<<dim3(gx, NH), 256, 0, stream>>>(cur, Ws[L], h_buf);
        score_kernel<<<(NH * NNODES + 255) / 256, 256, 0, stream>>>(h_buf, As[L], s_src, s_dst);
        aggregate_kernel<<<NNODES, 128, 0, stream>>>(h_buf, s_src, s_dst, neighbours, nxt);
        cur = nxt;
        nxt = (nxt == emb_a) ? emb_b : emb_a;
    }

    mean_nodes<<<1, 128, 0, stream>>>(cur, gvec);
    mlp_kernel<<<1, 128, 0, stream>>>(gvec,
                                      ln1_g, ln1_b, lin1_W,
                                      ln2_g, ln2_b, lin2_W, lin2_b,
                                      ln3_g, ln3_b, lin3_W, lin3_b,
                                      lin4_W, lin4_b,
                                      (float*)d_out);
}